// UpsampleLoss_80058190397996
// MI455X (gfx1250) — compile-verified
//
#include <hip/hip_runtime.h>
#include <math.h>

typedef __attribute__((ext_vector_type(2))) float v2f;
typedef __attribute__((ext_vector_type(8))) float v8f;

#define RADIUS_C   0.07f
#define H_BW2_INV  (1.0f / (0.03f * 0.03f))
#define EPS_C      1e-12f
#define SIGMA_INV  100.0f      // 1 / 0.01
#define FX         111
#define FY         62
#define MAXN       4096

__device__ __forceinline__ float wshfl(float x, int lane)  { return __shfl(x, lane, 32); }
__device__ __forceinline__ float wshflx(float x, int mask) { return __shfl_xor(x, mask, 32); }

// Async DMA a [16B-aligned, multiple-of-16B] region from global into LDS.
// GLOBAL_LOAD_ASYNC_TO_LDS_B128, GVS mode: saddr = base, vaddr = 32b offset,
// vdst = LDS byte offset (generic LDS addr truncates to allocation offset).
// Tracked by ASYNCcnt; caller still needs __syncthreads() for cross-wave vis.
__device__ __forceinline__ void async_stage16(const float* __restrict__ src,
                                              float* dst_lds, int nfloats,
                                              int tid, int nthreads) {
    const unsigned ldsBase = (unsigned)(uintptr_t)(void*)dst_lds;
    const int nchunks = nfloats >> 2;                 // 16-byte chunks
    for (int i = tid; i < nchunks; i += nthreads) {
        const unsigned loff = ldsBase + (unsigned)(i * 16);
        const unsigned goff = (unsigned)(i * 16);
        asm volatile("global_load_async_to_lds_b128 %0, %1, %2"
                     :: "v"(loff), "v"(goff), "s"(src) : "memory");
    }
    asm volatile("s_wait_asynccnt 0x0" ::: "memory");
}

// ---------------------------------------------------------------------------
// ws layout (floats):
//  [0..3]  pred: min_y, max_y, min_z, max_z   (coords 1 and 2)
//  [4..7]  gt  : min_y, max_y, min_z, max_z
//  [8..11] chamfer sums: [8+2*b+dir]  dir0 = "for" (gt rows), dir1 = "bac"
//  [12]    repulsion sum
//  [13]    frame sum of squared diffs
// ---------------------------------------------------------------------------

__global__ void init_kernel(float* ws) {
    int t = threadIdx.x;
    if (t < 6) ws[8 + t] = 0.0f;
}

// ----------------------- Chamfer via WMMA f32 16x16x4 ----------------------
// grid: (N/64, B, 2), block: 128 (4 waves, one 16-row tile per wave)
__global__ __launch_bounds__(128)
void cd_min_kernel(const float* __restrict__ pred, const float* __restrict__ gt,
                   float* __restrict__ acc, int N) {
    __shared__ __align__(16) float lcol[MAXN * 3];

    const int dir = blockIdx.z;
    const int b   = blockIdx.y;
    const float* rows = (dir == 0 ? gt   : pred) + (size_t)b * N * 3;
    const float* cols = (dir == 0 ? pred : gt)   + (size_t)b * N * 3;

    async_stage16(cols, lcol, N * 3, threadIdx.x, blockDim.x);
    __syncthreads();

    const int lane = threadIdx.x & 31;
    const int wave = threadIdx.x >> 5;
    const int m    = lane & 15;       // row (A) / col (B) index within tile
    const int half = lane >> 4;       // 0: holds (K0,K1)=(x,y); 1: (K2,K3)=(z,0)
    const int rowBase = (blockIdx.x * 4 + wave) << 4;

    // A fragment: 16 row points, K padded to 4 with 0
    const float* rp = rows + (size_t)(rowBase + m) * 3;
    const float r0 = rp[0], r1 = rp[1], r2 = rp[2];
    v2f a;
    a.x = half ? r2   : r0;
    a.y = half ? 0.0f : r1;
    const float a2row = r0 * r0 + r1 * r1 + r2 * r2;    // |row (lane&15)|^2
    float a2s[8];                                       // |row (half*8+v)|^2
#pragma unroll
    for (int v = 0; v < 8; ++v) a2s[v] = wshfl(a2row, (half << 3) + v);

    float mn[8];
#pragma unroll
    for (int v = 0; v < 8; ++v) mn[v] = 3.0e38f;

    for (int cb = 0; cb < N; cb += 16) {
        const float* cp = &lcol[(cb + m) * 3];
        const float c0 = cp[0], c1 = cp[1], c2 = cp[2];
        v2f bf;
        bf.x = half ? c2   : c0;
        bf.y = half ? 0.0f : c1;
        const float b2 = c0 * c0 + c1 * c1 + c2 * c2;   // |col (cb+lane&15)|^2

        v8f c = {};
        c = __builtin_amdgcn_wmma_f32_16x16x4_f32(false, a, false, bf,
                                                  (short)0, c, false, false);
#pragma unroll
        for (int v = 0; v < 8; ++v) {
            const float d2 = fmaxf(a2s[v] + b2 - 2.0f * c[v], 0.0f);
            mn[v] = fminf(mn[v], d2);
        }
    }

    // min across the 16 lanes of each half (rows 0..7 live in lanes 0..15,
    // rows 8..15 in lanes 16..31)
#pragma unroll
    for (int off = 8; off >= 1; off >>= 1) {
#pragma unroll
        for (int v = 0; v < 8; ++v) mn[v] = fminf(mn[v], wshflx(mn[v], off));
    }
    if (m == 0) {
        float s = 0.0f;
#pragma unroll
        for (int v = 0; v < 8; ++v) s += mn[v];
        atomicAdd(&acc[b * 2 + dir], s);
    }
}

// ----------------------------- Repulsion -----------------------------------
// grid: (N/256, B), block: 256. 5 smallest d2 per point (incl. self=0).
__global__ __launch_bounds__(256)
void rep_kernel(const float* __restrict__ pred, float* __restrict__ acc, int N) {
    __shared__ __align__(16) float lp[MAXN * 3];
    const float* P = pred + (size_t)blockIdx.y * N * 3;
    async_stage16(P, lp, N * 3, threadIdx.x, blockDim.x);
    __syncthreads();

    const int i = blockIdx.x * 256 + threadIdx.x;
    if (i >= N) return;

    const float px = lp[3 * i], py = lp[3 * i + 1], pz = lp[3 * i + 2];
    float nn[5];
#pragma unroll
    for (int k = 0; k < 5; ++k) nn[k] = 3.0e38f;

    for (int j = 0; j < N; ++j) {
        const float dx = lp[3 * j]     - px;
        const float dy = lp[3 * j + 1] - py;
        const float dz = lp[3 * j + 2] - pz;
        const float d2 = dx * dx + dy * dy + dz * dz;
        if (d2 < nn[4]) {
            nn[4] = d2;
#pragma unroll
            for (int k = 4; k > 0; --k) {
                if (nn[k] < nn[k - 1]) { float t = nn[k]; nn[k] = nn[k - 1]; nn[k - 1] = t; }
            }
        }
    }
    // nn[0] ~ 0 is self; neighbors are nn[1..4]
    float s = 0.0f;
#pragma unroll
    for (int k = 1; k < 5; ++k) {
        const float d2 = fmaxf(nn[k], EPS_C);
        const float d  = sqrtf(d2);
        s += (RADIUS_C - d) * __expf(-d2 * H_BW2_INV);
    }
    atomicAdd(acc, s);
}

// ------------------------ Frame: min/max reduction --------------------------
// 1 block, 256 threads; coords 1,2 over flattened B*N points, both clouds.
__global__ __launch_bounds__(256)
void minmax_kernel(const float* __restrict__ pred, const float* __restrict__ gt,
                   float* __restrict__ ws, int total) {
    float v[8];
    v[0] = v[2] = v[4] = v[6] =  3.0e38f;   // mins
    v[1] = v[3] = v[5] = v[7] = -3.0e38f;   // maxs
    for (int i = threadIdx.x; i < total; i += 256) {
        float y = pred[3 * i + 1], z = pred[3 * i + 2];
        v[0] = fminf(v[0], y); v[1] = fmaxf(v[1], y);
        v[2] = fminf(v[2], z); v[3] = fmaxf(v[3], z);
        y = gt[3 * i + 1]; z = gt[3 * i + 2];
        v[4] = fminf(v[4], y); v[5] = fmaxf(v[5], y);
        v[6] = fminf(v[6], z); v[7] = fmaxf(v[7], z);
    }
#pragma unroll
    for (int off = 16; off >= 1; off >>= 1) {
#pragma unroll
        for (int k = 0; k < 8; k += 2) v[k] = fminf(v[k], wshflx(v[k], off));
#pragma unroll
        for (int k = 1; k < 8; k += 2) v[k] = fmaxf(v[k], wshflx(v[k], off));
    }
    __shared__ float red[8][8];
    const int wave = threadIdx.x >> 5, lane = threadIdx.x & 31;
    if (lane == 0) {
#pragma unroll
        for (int k = 0; k < 8; ++k) red[wave][k] = v[k];
    }
    __syncthreads();
    if (threadIdx.x == 0) {
#pragma unroll
        for (int k = 0; k < 8; ++k) {
            float r = red[0][k];
            for (int w = 1; w < 8; ++w)
                r = (k & 1) ? fmaxf(r, red[w][k]) : fminf(r, red[w][k]);
            ws[k] = r;
        }
    }
}

// ------------------------ Frame splat via WMMA -------------------------------
// Each wave owns 16 grid cells (rows); points are columns (K=2 padded to 4).
// grid: ceil(6882/64) blocks, block: 128 (4 waves).
__global__ __launch_bounds__(128)
void frame_kernel(const float* __restrict__ pred, const float* __restrict__ gt,
                  const float* __restrict__ mm, float* __restrict__ acc, int total) {
    __shared__ __align__(16) float su[1024], sv[1024], tu[1024], tv[1024];

    const int lane = threadIdx.x & 31;
    const int wave = threadIdx.x >> 5;
    const int m    = lane & 15;
    const int half = lane >> 4;
    const int rowBase = blockIdx.x * 64 + wave * 16;

    const float mnyP = mm[0], syP = (float)(FX - 1) / (mm[1] - mm[0]);
    const float mnzP = mm[2], szP = (float)(FY - 1) / (mm[3] - mm[2]);
    const float mnyG = mm[4], syG = (float)(FX - 1) / (mm[5] - mm[4]);
    const float mnzG = mm[6], szG = (float)(FY - 1) / (mm[7] - mm[6]);

    // A fragment: 16 grid cells (cx, cy, 0, 0)
    const int c = rowBase + m;                 // may exceed FX*FY in tail tile
    const float cx = (float)(c / FY);
    const float cy = (float)(c % FY);
    v2f a;
    a.x = half ? 0.0f : cx;
    a.y = half ? 0.0f : cy;
    const float a2row = cx * cx + cy * cy;
    float a2s[8];
#pragma unroll
    for (int v = 0; v < 8; ++v) a2s[v] = wshfl(a2row, (half << 3) + v);

    float pf[8], gf[8];
#pragma unroll
    for (int v = 0; v < 8; ++v) { pf[v] = 0.0f; gf[v] = 0.0f; }

    for (int base = 0; base < total; base += 1024) {
        const int cnt = min(1024, total - base);
        const int cntA = (cnt + 15) & ~15;
        __syncthreads();
        for (int i = threadIdx.x; i < cnt; i += blockDim.x) {
            su[i] = (pred[(base + i) * 3 + 1] - mnyP) * syP;
            sv[i] = (pred[(base + i) * 3 + 2] - mnzP) * szP;
            tu[i] = (gt[(base + i) * 3 + 1] - mnyG) * syG;
            tv[i] = (gt[(base + i) * 3 + 2] - mnzG) * szG;
        }
        for (int i = cnt + threadIdx.x; i < cntA; i += blockDim.x) {
            su[i] = 1.0e9f; sv[i] = 1.0e9f; tu[i] = 1.0e9f; tv[i] = 1.0e9f;
        }
        __syncthreads();

        for (int cb = 0; cb < cntA; cb += 16) {
            // pred tile
            float u = su[cb + m], w = sv[cb + m];
            v2f bf;
            bf.x = half ? 0.0f : u;
            bf.y = half ? 0.0f : w;
            float b2 = u * u + w * w;
            v8f d = {};
            d = __builtin_amdgcn_wmma_f32_16x16x4_f32(false, a, false, bf,
                                                      (short)0, d, false, false);
#pragma unroll
            for (int v = 0; v < 8; ++v)
                pf[v] += __expf(-fmaxf(a2s[v] + b2 - 2.0f * d[v], 0.0f) * SIGMA_INV);

            // gt tile
            u = tu[cb + m]; w = tv[cb + m];
            bf.x = half ? 0.0f : u;
            bf.y = half ? 0.0f : w;
            b2 = u * u + w * w;
            v8f e = {};
            e = __builtin_amdgcn_wmma_f32_16x16x4_f32(false, a, false, bf,
                                                      (short)0, e, false, false);
#pragma unroll
            for (int v = 0; v < 8; ++v)
                gf[v] += __expf(-fmaxf(a2s[v] + b2 - 2.0f * e[v], 0.0f) * SIGMA_INV);
        }
    }

    // sum across the 16 lanes of each half
#pragma unroll
    for (int off = 8; off >= 1; off >>= 1) {
#pragma unroll
        for (int v = 0; v < 8; ++v) {
            pf[v] += wshflx(pf[v], off);
            gf[v] += wshflx(gf[v], off);
        }
    }
    if (m == 0) {
        float s = 0.0f;
#pragma unroll
        for (int v = 0; v < 8; ++v) {
            const int row = rowBase + (half << 3) + v;
            if (row < FX * FY) {
                const float d = pf[v] - gf[v];
                s += d * d;
            }
        }
        atomicAdd(acc, s);
    }
}

// ------------------------------- Finalize -----------------------------------
__global__ void finalize_kernel(const float* __restrict__ pcd_radius,
                                const float* __restrict__ ws,
                                float* __restrict__ out, int B, int N) {
    if (threadIdx.x != 0 || blockIdx.x != 0) return;
    float cd = 0.0f;
    for (int b = 0; b < B; ++b)
        cd += (0.8f * ws[8 + 2 * b] + 0.2f * ws[8 + 2 * b + 1]) / pcd_radius[b];
    out[0] = 100.0f * cd / (float)(B * N);
    out[1] = ws[12] / (float)(B * N * 4);
    out[2] = ws[13] / (float)(FX * FY);
}

// -----------------------------------------------------------------------------
extern "C" void kernel_launch(void* const* d_in, const int* in_sizes, int n_in,
                              void* d_out, int out_size, void* d_ws, size_t ws_size,
                              hipStream_t stream) {
    (void)n_in; (void)out_size; (void)ws_size;
    const float* pred   = (const float*)d_in[0];
    const float* gt     = (const float*)d_in[1];
    const float* radius = (const float*)d_in[2];
    const int B = in_sizes[2];               // pcd_radius has B elements
    const int N = in_sizes[0] / (3 * B);
    float* ws  = (float*)d_ws;
    float* out = (float*)d_out;

    init_kernel<<<1, 32, 0, stream>>>(ws);

    dim3 cdGrid(N / 64, B, 2);
    cd_min_kernel<<<cdGrid, 128, 0, stream>>>(pred, gt, ws + 8, N);

    dim3 repGrid(N / 256, B);
    rep_kernel<<<repGrid, 256, 0, stream>>>(pred, ws + 12, N);

    minmax_kernel<<<1, 256, 0, stream>>>(pred, gt, ws, B * N);

    frame_kernel<<<(FX * FY + 63) / 64, 128, 0, stream>>>(pred, gt, ws, ws + 13, B * N);

    finalize_kernel<<<1, 1, 0, stream>>>(radius, ws, out, B, N);
}